// FixedAdditivePositionalBias_28690381537578
// MI455X (gfx1250) — compile-verified
//
#include <hip/hip_runtime.h>

// out[i] = W[idx[i] - 1]  -- pure gather, HBM-bandwidth bound (~26 MB total).
// CDNA5 path: stage the 200-entry table into LDS with the async global->LDS
// DMA (ASYNCcnt), then vectorized b128 index loads + LDS gathers + b128 stores.

#define MAX_RANKS 200
#define BLOCK 256

__global__ __launch_bounds__(BLOCK) void pos_bias_gather_kernel(
    const int* __restrict__ idx,
    const float* __restrict__ W,
    float* __restrict__ out,
    int total,        // total scalar elements (16384*200)
    int total_vec)    // total/4
{
    __shared__ float w_lds[MAX_RANKS];
    const int tid = threadIdx.x;

    // ---- Stage W (800 B) into LDS via CDNA5 async global->LDS copy ----
    // global_load_async_to_lds_b32: VDST = per-lane LDS byte address,
    // VADDR = per-lane 64-bit global address. Tracked by ASYNCcnt.
    if (tid < MAX_RANKS) {
        // Low 32 bits of a flat LDS pointer are the LDS byte offset (ISA 10.2
        // aperture mapping).
        unsigned lds_addr = (unsigned)(size_t)(&w_lds[tid]);
        const float* gptr = W + tid;
        asm volatile("global_load_async_to_lds_b32 %0, %1, off"
                     :
                     : "v"(lds_addr), "v"(gptr)
                     : "memory");
    }
    // Drain this wave's async transfers, then make LDS visible workgroup-wide.
    asm volatile("s_wait_asynccnt 0" ::: "memory");
    __syncthreads();

    // ---- Vectorized gather: 4 elements per thread ----
    const int gid = blockIdx.x * BLOCK + tid;
    if (gid < total_vec) {
        const int4 v = ((const int4*)idx)[gid];   // global_load_b128
        float4 r;
        r.x = w_lds[v.x - 1];                     // ds_load_b32 x4
        r.y = w_lds[v.y - 1];
        r.z = w_lds[v.z - 1];
        r.w = w_lds[v.w - 1];
        ((float4*)out)[gid] = r;                  // global_store_b128
    } else if (gid == total_vec) {
        // Scalar tail (not hit for 3,276,800 % 4 == 0, kept for safety).
        for (int i = total_vec * 4; i < total; ++i) {
            out[i] = w_lds[idx[i] - 1];
        }
    }
}

extern "C" void kernel_launch(void* const* d_in, const int* in_sizes, int n_in,
                              void* d_out, int out_size, void* d_ws, size_t ws_size,
                              hipStream_t stream) {
    (void)n_in; (void)out_size; (void)d_ws; (void)ws_size;

    const int*   idx = (const int*)d_in[0];   // "inputs": [16384, 200] int
    const float* W   = (const float*)d_in[1]; // "W": [200, 1] float32
    float*       out = (float*)d_out;         // [16384, 200, 1] float32

    const int total     = in_sizes[0];        // 3,276,800
    const int total_vec = total / 4;          // 819,200
    // One vec4 per thread; +1 block headroom covers the (empty) scalar tail.
    const int blocks = (total_vec + BLOCK - 1) / BLOCK + ((total & 3) ? 1 : 0);

    pos_bias_gather_kernel<<<blocks, BLOCK, 0, stream>>>(idx, W, out, total, total_vec);
}